// ChannelRegMixer_1752346656902
// MI455X (gfx1250) — compile-verified
//
#include <hip/hip_runtime.h>
#include <cstdint>

#define EPSV 1e-6f
#define STEPV 0.25f
#define KAPPA 0.03f
#define BASE_LAMBDA 0.03f

#define Bn 8
#define Cn 64
#define Hn 256
#define Wn 256
#define NIMG (Bn*Cn)          /* 512 */
#define NPIX (Hn*Wn)          /* 65536 */

#define TILE 32
#define TX (TILE+4)           /* 36: tile + halo-2 */
#define BXh (TILE+2)          /* 34: tile + halo-1 */

typedef float v2f __attribute__((ext_vector_type(2)));
typedef float v8f __attribute__((ext_vector_type(8)));
typedef unsigned int uint4v __attribute__((ext_vector_type(4)));
typedef int int4v  __attribute__((ext_vector_type(4)));
typedef int int8v  __attribute__((ext_vector_type(8)));

// ---------------------------------------------------------------------------
// Kernel 1: per-(b,c) statistics. One block per image; thread j owns column j
// (W==256==blockDim.x), vertical register reuse -> 3 loads/pixel, L2-resident.
// ---------------------------------------------------------------------------
__global__ __launch_bounds__(256) void stats_kernel(const float* __restrict__ x,
                                                    float* __restrict__ stats) {
  const int bc = blockIdx.x;
  const int j  = threadIdx.x;
  const float* img = x + (size_t)bc * NPIX;

  float sum = 0.f, sumsq = 0.f, mx = -INFINITY;
  float s_agx = 0.f, s_agy = 0.f, s_lap = 0.f, s_gx2 = 0.f, s_gy2 = 0.f;

  float cur = img[j];     // row 0
  float up  = 0.0f;       // zero pad above
  for (int i = 0; i < Hn; ++i) {
    const float* row = img + i * Wn;
    float xl = (j > 0)      ? row[j - 1] : 0.0f;
    float xr = (j < Wn - 1) ? row[j + 1] : 0.0f;
    float dn = (i < Hn - 1) ? img[(i + 1) * Wn + j] : 0.0f;

    sum   += cur;
    sumsq += cur * cur;
    mx     = fmaxf(mx, cur);
    if (j < Wn - 1) { float g = xr - cur; s_agx += fabsf(g); s_gx2 += g * g; }
    if (i < Hn - 1) { float g = dn - cur; s_agy += fabsf(g); s_gy2 += g * g; }
    float lap = xl + xr + up + dn - 4.0f * cur;   // zero padding
    s_lap += fabsf(lap);

    up = cur; cur = dn;
  }

  __shared__ float red[256];
  float vals[8] = {sum, mx, s_agx, s_agy, s_lap, sumsq, s_gx2, s_gy2};
  float res[8];
  for (int q = 0; q < 8; ++q) {
    red[j] = vals[q];
    __syncthreads();
    for (int off = 128; off > 0; off >>= 1) {
      if (j < off) {
        float o = red[j + off];
        red[j] = (q == 1) ? fmaxf(red[j], o) : (red[j] + o);
      }
      __syncthreads();
    }
    if (j == 0) res[q] = red[0];
    __syncthreads();
  }

  if (j == 0) {
    const float invN = 1.0f / (float)NPIX;
    const float nGx = (float)(Hn * (Wn - 1));
    const float nGy = (float)((Hn - 1) * Wn);
    float gap   = res[0] * invN;
    float gmp   = res[1];
    float gmean = 0.5f * (res[2] / nGx + res[3] / nGy);
    float lmean = res[4] * invN;
    float var   = res[5] * invN - gap * gap;
    float A  = res[6] / nGx;
    float Bv = res[7] / nGy;
    float aniso = fabsf(A - Bv) / (A + Bv + 1e-6f);
    float* s = stats + bc * 6;
    s[0] = gap; s[1] = gmp; s[2] = gmean; s[3] = lmean; s[4] = var; s[5] = aniso;
  }
}

// ---------------------------------------------------------------------------
// Kernel 2: head GEMM [512,6]x[6,4] via V_WMMA_F32_16X16X4_F32 (K padded to 8,
// two WMMAs), then softmax(3)+softplus. One wave per 16 rows -> 32 blocks.
// A-frag (16x4 f32): lane m (0-15) row M=m holds K=0,1; lane m+16 holds K=2,3.
// B-frag (4x16 f32): VGPR0 = K0 (lanes 0-15)/K2 (16-31), VGPR1 = K1/K3, N=lane&15.
// ---------------------------------------------------------------------------
__global__ __launch_bounds__(32) void head_kernel(const float* __restrict__ stats,
                                                  const float* __restrict__ mlp_w,  // [4][6]
                                                  const float* __restrict__ mlp_b,  // [4]
                                                  float* __restrict__ mix_out,      // [512][3]
                                                  float* __restrict__ lam_out) {    // [512]
  const int lane = threadIdx.x;
  const int half = lane >> 4;
  const int m    = lane & 15;
  const int row  = blockIdx.x * 16 + m;
  const float* s = stats + row * 6;

  v2f a1, a2;
  a1.x = half ? s[2] : s[0];
  a1.y = half ? s[3] : s[1];
  a2.x = half ? 0.0f : s[4];     // K=4..7: features 6,7 are zero pad
  a2.y = half ? 0.0f : s[5];

  const int n = m;               // output column
  v2f b1 = {0.0f, 0.0f}, b2 = {0.0f, 0.0f};
  if (n < 4) {
    b1.x = mlp_w[n * 6 + (half ? 2 : 0)];
    b1.y = mlp_w[n * 6 + (half ? 3 : 1)];
    if (!half) { b2.x = mlp_w[n * 6 + 4]; b2.y = mlp_w[n * 6 + 5]; }
  }

  v8f acc = {};
  acc = __builtin_amdgcn_wmma_f32_16x16x4_f32(false, a1, false, b1, (short)0, acc, false, false);
  acc = __builtin_amdgcn_wmma_f32_16x16x4_f32(false, a2, false, b2, (short)0, acc, false, false);

  // D layout: VGPR v -> M=v (lanes 0-15) / M=v+8 (lanes 16-31), N = lane&15
  __shared__ float hd[16][4];
  if (n < 4) {
    float bias = mlp_b[n];
#pragma unroll
    for (int v = 0; v < 8; ++v) hd[v + half * 8][n] = acc[v] + bias;
  }
  __syncthreads();

  if (lane < 16) {
    const int R = blockIdx.x * 16 + lane;
    float h0 = hd[lane][0], h1 = hd[lane][1], h2 = hd[lane][2], h3 = hd[lane][3];
    float mM = fmaxf(h0, fmaxf(h1, h2));
    float e0 = expf(h0 - mM), e1 = expf(h1 - mM), e2 = expf(h2 - mM);
    float inv = 1.0f / (e0 + e1 + e2);
    mix_out[R * 3 + 0] = e0 * inv;
    mix_out[R * 3 + 1] = e1 * inv;
    mix_out[R * 3 + 2] = e2 * inv;
    float sp = (h3 > 20.0f) ? h3 : log1pf(expf(h3));
    lam_out[R] = BASE_LAMBDA * sp;
  }
}

// ---------------------------------------------------------------------------
// Kernel 3: gate conv + two fused diffusion steps, tiled 32x32 with halo-2.
// Interior tiles stream x into LDS with the Tensor Data Mover (async DMA,
// TENSORcnt); edge tiles use guarded loads with zero fill (== zero padding).
// ---------------------------------------------------------------------------
__device__ __forceinline__ float qtv(float g) { return g / (fabsf(g) + EPSV); }
__device__ __forceinline__ float qad(float g) {
  float ag = fabsf(g);
  float t  = ag * (1.0f / KAPPA);
  float w  = 1.0f / (1.0f + t * t);
  return w * g / (ag + EPSV);
}

__global__ __launch_bounds__(256) void fused_kernel(const float* __restrict__ x,
                                                    const float* __restrict__ gate_w,  // [64][9]
                                                    const float* __restrict__ gate_b,  // [64]
                                                    const float* __restrict__ mix,     // [512][3]
                                                    const float* __restrict__ lam,     // [512]
                                                    float* __restrict__ out) {
  __shared__ float tile[TX][TX];    // x with halo-2
  __shared__ float lamf[BXh][BXh];  // lambda * gate, halo-1
  __shared__ float buf1[BXh][BXh];  // step-1 result, halo-1
  __shared__ float sc[14];          // gw[9], gb, w0,w1,w2, lam

  const int bc  = blockIdx.z;
  const int c   = bc % Cn;
  const int gx0 = blockIdx.x * TILE - 2;
  const int gy0 = blockIdx.y * TILE - 2;
  const float* img = x + (size_t)bc * NPIX;
  const int tid = threadIdx.x;

  if (tid < 9)        sc[tid] = gate_w[c * 9 + tid];
  else if (tid == 9)  sc[9]  = gate_b[c];
  else if (tid < 13)  sc[tid] = mix[bc * 3 + (tid - 10)];
  else if (tid == 13) sc[13] = lam[bc];

  const bool interior = (gx0 >= 0) && (gy0 >= 0) && (gx0 + TX <= Wn) && (gy0 + TX <= Hn);
  if (interior) {
    if (tid < 32) {  // wave 0 issues the TDM descriptor (per-wave instruction)
      unsigned ldsOff = (unsigned)(uintptr_t)(&tile[0][0]);
      uint64_t ga = (uint64_t)(uintptr_t)(img + (size_t)gy0 * Wn + gx0);
      unsigned rem0 = (unsigned)(Wn - gx0);           // remaining extent for OOB->0
      unsigned rem1 = (unsigned)(Hn - gy0);
      uint64_t str0 = (uint64_t)Wn;
      uint64_t str1 = (uint64_t)Wn * (uint64_t)Hn;
      unsigned d[12];
      d[0]  = 1u;                                                  // count=1 user D#
      d[1]  = ldsOff;                                              // lds_addr
      d[2]  = (unsigned)(ga & 0xFFFFFFFFu);                        // global_addr lo
      d[3]  = (unsigned)((ga >> 32) & 0x1FFFFFFu) | (2u << 30);    // addr hi | type=2
      d[4]  = (2u << 16);                                          // data_size=4B
      d[5]  = (rem0 & 0xFFFFu) << 16;                              // tensor_dim0 lo16
      d[6]  = (rem0 >> 16) | ((rem1 & 0xFFFFu) << 16);             // dim0 hi | dim1 lo
      d[7]  = (rem1 >> 16) | ((unsigned)TX << 16);                 // dim1 hi | tile_dim0
      d[8]  = (unsigned)TX;                                        // tile_dim1 (dim2=0)
      d[9]  = (unsigned)(str0 & 0xFFFFFFFFu);                      // stride0 lo
      d[10] = (unsigned)((str0 >> 32) & 0xFFFFu) | ((unsigned)(str1 & 0xFFFFu) << 16);
      d[11] = (unsigned)(str1 >> 16);                              // stride1 hi32
#pragma unroll
      for (int k = 0; k < 12; ++k)
        d[k] = (unsigned)__builtin_amdgcn_readfirstlane((int)d[k]);
      uint4v g0 = {d[0], d[1], d[2], d[3]};
      int8v  g1 = {(int)d[4], (int)d[5], (int)d[6], (int)d[7],
                   (int)d[8], (int)d[9], (int)d[10], (int)d[11]};
      int4v  z4 = {0, 0, 0, 0};
      int8v  z8 = {0, 0, 0, 0, 0, 0, 0, 0};
      __builtin_amdgcn_tensor_load_to_lds(g0, g1, z4, z4, z8, 0);
      __builtin_amdgcn_s_wait_tensorcnt(0);
    }
  } else {
    for (int idx = tid; idx < TX * TX; idx += blockDim.x) {
      int ty = idx / TX, txx = idx % TX;
      int gi = gy0 + ty, gj = gx0 + txx;
      float v = 0.0f;
      if (gi >= 0 && gi < Hn && gj >= 0 && gj < Wn) v = img[gi * Wn + gj];
      tile[ty][txx] = v;
    }
  }
  __syncthreads();

  const float w0 = sc[10], w1 = sc[11], w2 = sc[12], lam_bc = sc[13];

  // Phase A: gate conv + lambda field + diffusion step 1 on 34x34 (halo-1)
  for (int idx = tid; idx < BXh * BXh; idx += blockDim.x) {
    int i1 = idx / BXh, j1 = idx % BXh;
    int ti = i1 + 1, tj = j1 + 1;       // tile coords
    int gi = gy0 + ti, gj = gx0 + tj;   // global coords

    float conv = sc[9];
#pragma unroll
    for (int u = 0; u < 3; ++u)
#pragma unroll
      for (int v = 0; v < 3; ++v)
        conv += sc[u * 3 + v] * tile[ti - 1 + u][tj - 1 + v];
    float g  = 1.0f / (1.0f + expf(-conv));
    float lf = lam_bc * g;
    lamf[i1][j1] = lf;

    float o = 0.0f;
    if (gi >= 0 && gi < Hn && gj >= 0 && gj < Wn) {
      float cen = tile[ti][tj];
      float lft = tile[ti][tj - 1], rgt = tile[ti][tj + 1];
      float up  = tile[ti - 1][tj], dn  = tile[ti + 1][tj];
      bool hL = gj > 0, hR = gj < Wn - 1, hU = gi > 0, hD = gi < Hn - 1;
      float gxp = rgt - cen, gxm = cen - lft, gyp = dn - cen, gym = cen - up;
      float tvd  = (hR ? qtv(gxp) : 0.0f) - (hL ? qtv(gxm) : 0.0f)
                 + (hD ? qtv(gyp) : 0.0f) - (hU ? qtv(gym) : 0.0f);
      float add_ = (hR ? qad(gxp) : 0.0f) - (hL ? qad(gxm) : 0.0f)
                 + (hD ? qad(gyp) : 0.0f) - (hU ? qad(gym) : 0.0f);
      float lap = lft + rgt + up + dn - 4.0f * cen;  // zero padding via LDS zeros
      o = cen - STEPV * lf * (w0 * tvd + w1 * lap + w2 * add_);
    }
    buf1[i1][j1] = o;
  }
  __syncthreads();

  // Phase B: diffusion step 2 on the 32x32 interior, write out
  float* oimg = out + (size_t)bc * NPIX;
  for (int idx = tid; idx < TILE * TILE; idx += blockDim.x) {
    int i2 = idx / TILE, j2 = idx % TILE;
    int bi = i2 + 1, bj = j2 + 1;
    int gi = gy0 + 2 + i2, gj = gx0 + 2 + j2;

    float cen = buf1[bi][bj];
    float lft = buf1[bi][bj - 1], rgt = buf1[bi][bj + 1];
    float up  = buf1[bi - 1][bj], dn  = buf1[bi + 1][bj];
    bool hL = gj > 0, hR = gj < Wn - 1, hU = gi > 0, hD = gi < Hn - 1;
    float gxp = rgt - cen, gxm = cen - lft, gyp = dn - cen, gym = cen - up;
    float tvd  = (hR ? qtv(gxp) : 0.0f) - (hL ? qtv(gxm) : 0.0f)
               + (hD ? qtv(gyp) : 0.0f) - (hU ? qtv(gym) : 0.0f);
    float add_ = (hR ? qad(gxp) : 0.0f) - (hL ? qad(gxm) : 0.0f)
               + (hD ? qad(gyp) : 0.0f) - (hU ? qad(gym) : 0.0f);
    float lap = lft + rgt + up + dn - 4.0f * cen;
    float o = cen - STEPV * lamf[bi][bj] * (w0 * tvd + w1 * lap + w2 * add_);
    oimg[gi * Wn + gj] = o;
  }
}

// ---------------------------------------------------------------------------
extern "C" void kernel_launch(void* const* d_in, const int* in_sizes, int n_in,
                              void* d_out, int out_size, void* d_ws, size_t ws_size,
                              hipStream_t stream) {
  const float* x      = (const float*)d_in[0];
  const float* mlp_w  = (const float*)d_in[1];
  const float* mlp_b  = (const float*)d_in[2];
  const float* gate_w = (const float*)d_in[3];
  const float* gate_b = (const float*)d_in[4];

  float* out = (float*)d_out;                       // [B,C,H,W]
  float* mix = out + (size_t)NIMG * NPIX;           // [B,C,3]
  float* lamv = mix + (size_t)NIMG * 3;             // [B,C]
  float* stats = (float*)d_ws;                      // [512][6]

  stats_kernel<<<NIMG, 256, 0, stream>>>(x, stats);
  head_kernel<<<NIMG / 16, 32, 0, stream>>>(stats, mlp_w, mlp_b, mix, lamv);
  dim3 grid(Wn / TILE, Hn / TILE, NIMG);
  fused_kernel<<<grid, 256, 0, stream>>>(x, gate_w, gate_b, mix, lamv, out);
}